// WatershedGroups_74191265071458
// MI455X (gfx1250) — compile-verified
//
#include <hip/hip_runtime.h>
#include <math.h>
#include <stdint.h>

// ---------------- problem constants (match reference) ----------------
#define B_    8
#define C_    3
#define H_    256
#define W_    256
#define HW_   (H_ * W_)
#define N_    (B_ * HW_)
#define G_    16          // MAX_GROUPS
#define RAD   20          // MIN_DIST -> window radius (win = 41)
#define BIGV  1.0e4f
#define FINF  3.0e38f

#ifndef __has_builtin
#define __has_builtin(x) 0
#endif

// CDNA5 WMMA: V_WMMA_F32_16X16X4_F32 (probe doc: 8 args for 16x16x4)
#if __has_builtin(__builtin_amdgcn_wmma_f32_16x16x4_f32)
#define HAVE_WMMA4 1
#else
#define HAVE_WMMA4 0
#endif

// CDNA5 async global->LDS copy (GLOBAL_LOAD_ASYNC_TO_LDS_B32, ASYNCcnt)
#if __has_builtin(__builtin_amdgcn_global_load_async_to_lds_b32) && \
    __has_builtin(__builtin_amdgcn_s_wait_asynccnt)
#define HAVE_ASYNC 1
#else
#define HAVE_ASYNC 0
#endif

typedef __attribute__((ext_vector_type(2))) float v2f;
typedef __attribute__((ext_vector_type(8))) float v8f;

// stage one f32 per lane into LDS (async path emits gfx1250 async-tensor ops)
// builtin signature (from hipcc diagnostic): (AS1 int*, AS3 int*, imm, imm)
__device__ __forceinline__ void stage_to_lds(const float* g, float* l) {
#if HAVE_ASYNC
  __builtin_amdgcn_global_load_async_to_lds_b32(
      (__attribute__((address_space(1))) int*)(int*)g,
      (__attribute__((address_space(3))) int*)(int*)l, 0, 0);
  __builtin_amdgcn_s_wait_asynccnt(0);
#else
  *l = *g;
#endif
}

// ---------------------------------------------------------------------
// K1: foreground mask. mean over 3 channels via V_WMMA_F32_16X16X4_F32.
//     A (16x4) = 1/3 weights, B (4x16) = channels of 16 pixels.
//     fg = (uint8)(mean*255) > 0  <=>  mean*255 >= 1
// ---------------------------------------------------------------------
__global__ void k_fg_wmma(const float* __restrict__ x, int* __restrict__ fg) {
  const int lane  = threadIdx.x & 31;
  const int wave  = (blockIdx.x * blockDim.x + threadIdx.x) >> 5;
  const int nwave = (gridDim.x * blockDim.x) >> 5;
  const int ngroups = N_ / 16;

  for (int g = wave; g < ngroups; g += nwave) {   // wave-uniform loop: EXEC all 1s
    const int p   = g * 16 + (lane & 15);
    const int b   = p / HW_;
    const int off = p - b * HW_;
    const float* base = x + (size_t)b * C_ * HW_ + off;

    v2f a, bb;
    if (lane < 16) {                 // (vgpr0,vgpr1) x lanes 0-15 = K0,K1
      a.x = 1.0f / 3.0f; a.y = 1.0f / 3.0f;
      bb.x = base[0];                // channel 0 -> K0
      bb.y = base[HW_];              // channel 1 -> K1
    } else {                         // lanes 16-31 = K2,K3
      a.x = 1.0f / 3.0f; a.y = 0.0f;
      bb.x = base[2 * HW_];          // channel 2 -> K2
      bb.y = 0.0f;                   // K3 = 0
    }

#if HAVE_WMMA4
    v8f c = {};
    // D = A x B + C ; every row of D holds the channel mean of pixel N
    c = __builtin_amdgcn_wmma_f32_16x16x4_f32(false, a, false, bb,
                                              (short)0, c, false, false);
    float mean = c[0];               // lane<16: D[0][lane]; lane>=16: D[8][lane-16]
#else
    float mean = 0.0f;
    if (lane < 16) mean = (bb.x + bb.y + base[2 * HW_]) * (1.0f / 3.0f);
#endif
    if (lane < 16) fg[p] = (mean * 255.0f >= 1.0f) ? 1 : 0;
  }
}

// ---------------------------------------------------------------------
// K2: 1D row distance (forward + backward scan along x), init BIG.
// ---------------------------------------------------------------------
__global__ void k_rowdist(const int* __restrict__ fg, float* __restrict__ rowd) {
  int t = blockIdx.x * blockDim.x + threadIdx.x;
  if (t >= B_ * H_) return;
  int b = t / H_, y = t - b * H_;
  const int* fr = fg + b * HW_ + y * W_;
  float* rr = rowd + b * HW_ + y * W_;
  float d = BIGV;
  for (int xx = 0; xx < W_; ++xx) { d = fr[xx] ? d + 1.0f : 0.0f; rr[xx] = d; }
  d = BIGV;
  for (int xx = W_ - 1; xx >= 0; --xx) {
    d = fr[xx] ? d + 1.0f : 0.0f;
    rr[xx] = fminf(rr[xx], d);
  }
}

// ---------------------------------------------------------------------
// K3: exact EDT column envelope: dist(y,x) = sqrt(min_y' f(y',x)^2 + (y-y')^2)
//     One block per (b,x) column; column staged to LDS (async on CDNA5).
// ---------------------------------------------------------------------
__global__ void k_edt(const float* __restrict__ rowd, float* __restrict__ dist) {
  __shared__ float s[H_];
  const int bx = blockIdx.x;
  const int b  = bx / W_;
  const int x  = bx - b * W_;
  const int y  = threadIdx.x;
  const int idx = b * HW_ + y * W_ + x;

  stage_to_lds(rowd + idx, &s[y]);
  __syncthreads();
  float fv = s[y];
  s[y] = fv * fv;                    // square in place (own slot only)
  __syncthreads();

  float best = FINF;
  const float fy = (float)y;
#pragma unroll 8
  for (int yp = 0; yp < H_; ++yp) {
    float d = fy - (float)yp;
    best = fminf(best, fmaf(d, d, s[yp]));
  }
  dist[idx] = sqrtf(best);
}

// ---------------------------------------------------------------------
// K4: separable 41x41 window max, row direction (out-of-range = -inf).
// ---------------------------------------------------------------------
__global__ void k_rowmax(const float* __restrict__ dist, float* __restrict__ mpr) {
  __shared__ float s[W_];
  const int br = blockIdx.x;
  const int b = br / H_, y = br - b * H_;
  const int x = threadIdx.x;
  const int idx = b * HW_ + y * W_ + x;
  s[x] = dist[idx];
  __syncthreads();
  int lo = x - RAD; if (lo < 0) lo = 0;
  int hi = x + RAD; if (hi > W_ - 1) hi = W_ - 1;
  float m = -FINF;
  for (int i = lo; i <= hi; ++i) m = fmaxf(m, s[i]);
  mpr[idx] = m;
}

// ---------------------------------------------------------------------
// K5: column direction window max + peak mask.
// ---------------------------------------------------------------------
__global__ void k_colmax_peaks(const float* __restrict__ mpr,
                               const float* __restrict__ dist,
                               const int* __restrict__ fg,
                               int* __restrict__ peaks) {
  __shared__ float s[H_];
  const int bx = blockIdx.x;
  const int b = bx / W_, x = bx - b * W_;
  const int y = threadIdx.x;
  const int idx = b * HW_ + y * W_ + x;

  stage_to_lds(mpr + idx, &s[y]);
  __syncthreads();

  int lo = y - RAD; if (lo < 0) lo = 0;
  int hi = y + RAD; if (hi > H_ - 1) hi = H_ - 1;
  float m = -FINF;
  for (int i = lo; i <= hi; ++i) m = fmaxf(m, s[i]);

  float dv = dist[idx];
  peaks[idx] = (fg[idx] && dv > 0.0f && dv >= m) ? 1 : 0;
}

// ---------------------------------------------------------------------
// K6: raster-order peak labels (cumsum) + JFA seed init. 1 block / image.
// ---------------------------------------------------------------------
__global__ void k_label(const int* __restrict__ peaks, int* __restrict__ lab,
                        float* __restrict__ sy, float* __restrict__ sx,
                        int* __restrict__ val, int* __restrict__ dmax) {
  __shared__ int cnt[256];
  const int b = blockIdx.x;
  const int t = threadIdx.x;
  const int* pk = peaks + b * HW_;
  const int base = t * 256;

  int local = 0;
  for (int i = 0; i < 256; ++i) local += pk[base + i];
  cnt[t] = local;
  __syncthreads();
  for (int o = 1; o < 256; o <<= 1) {          // Hillis-Steele inclusive scan
    int v = (t >= o) ? cnt[t - o] : 0;
    __syncthreads();
    cnt[t] += v;
    __syncthreads();
  }
  int run = cnt[t] - local;                    // exclusive prefix
  for (int i = 0; i < 256; ++i) {
    const int li = base + i;
    const int pv = pk[li];
    run += pv;
    const int gp = b * HW_ + li;
    const int y = li / W_;
    const int x = li - y * W_;
    lab[gp] = pv ? run : 0;
    sy[gp]  = pv ? (float)y : BIGV;
    sx[gp]  = pv ? (float)x : BIGV;
    val[gp] = pv;
  }
  if (t == 0) dmax[b] = 0;
}

// ---------------------------------------------------------------------
// K7: one jump-flood step (read src set, write dst set).
// ---------------------------------------------------------------------
__global__ void k_jfa(const float* __restrict__ ssy, const float* __restrict__ ssx,
                      const int* __restrict__ slab, const int* __restrict__ sval,
                      float* __restrict__ dsy, float* __restrict__ dsx,
                      int* __restrict__ dlab, int* __restrict__ dval, int step) {
  const int p = blockIdx.x * blockDim.x + threadIdx.x;
  if (p >= N_) return;
  const int b = p / HW_;
  const int off = p - b * HW_;
  const int y = off / W_;
  const int x = off - y * W_;
  const float fy = (float)y, fx = (float)x;

  float bsy = ssy[p], bsx = ssx[p];
  int blab = slab[p], bval = sval[p];
  float cur = bval ? ((fy - bsy) * (fy - bsy) + (fx - bsx) * (fx - bsx)) : FINF;

  const int dys[8] = {-1, -1, -1, 0, 0, 1, 1, 1};
  const int dxs[8] = {-1,  0,  1, -1, 1, -1, 0, 1};
#pragma unroll
  for (int i = 0; i < 8; ++i) {
    const int ny = y - dys[i] * step;
    const int nx = x - dxs[i] * step;
    if (ny < 0 || ny >= H_ || nx < 0 || nx >= W_) continue;
    const int q = b * HW_ + ny * W_ + nx;
    if (!sval[q]) continue;
    const float qy = ssy[q], qx = ssx[q];
    const float cd = (fy - qy) * (fy - qy) + (fx - qx) * (fx - qx);
    if (cd < cur) { cur = cd; bsy = qy; bsx = qx; blab = slab[q]; bval = 1; }
  }
  dsy[p] = bsy; dsx[p] = bsx; dlab[p] = blab; dval[p] = bval;
}

// ---------------------------------------------------------------------
// K8: segs = fg ? lab : 0 ; per-image max via atomics.
// ---------------------------------------------------------------------
__global__ void k_segs(const int* __restrict__ fg, const int* __restrict__ lab,
                       int* __restrict__ segs, int* __restrict__ dmax) {
  const int p = blockIdx.x * blockDim.x + threadIdx.x;
  if (p >= N_) return;
  const int s = fg[p] ? lab[p] : 0;
  segs[p] = s;
  atomicMax(&dmax[p / HW_], s);
}

// ---------------------------------------------------------------------
// K9: cap group count + one-hot to [B,G,H,W] float32.
// ---------------------------------------------------------------------
__global__ void k_onehot(const int* __restrict__ segs, const int* __restrict__ dmax,
                         float* __restrict__ out) {
  const int o = blockIdx.x * blockDim.x + threadIdx.x;
  if (o >= B_ * G_ * HW_) return;
  const int b = o / (G_ * HW_);
  const int r = o - b * (G_ * HW_);
  const int g = r / HW_;
  const int off = r - g * HW_;
  const int s = segs[b * HW_ + off];
  const float m = (float)dmax[b] + 1.0f;
  const float dv = (m >= (float)G_) ? (m / (float)G_) : 1.0f;
  const int sg = (int)floorf((float)s / dv);
  out[o] = (sg == g) ? 1.0f : 0.0f;
}

// ---------------------------------------------------------------------
extern "C" void kernel_launch(void* const* d_in, const int* in_sizes, int n_in,
                              void* d_out, int out_size, void* d_ws, size_t ws_size,
                              hipStream_t stream) {
  (void)in_sizes; (void)n_in; (void)out_size; (void)ws_size;
  const float* x = (const float*)d_in[0];
  float* out = (float*)d_out;

  char* ws = (char*)d_ws;
  const size_t SL = (size_t)N_ * sizeof(float);      // 2 MB per slot
  int*   fg    = (int*)  (ws + 0 * SL);
  float* rowd  = (float*)(ws + 1 * SL);              // -> reused as peaks
  float* dist  = (float*)(ws + 2 * SL);              // -> reused as lab_b
  float* mpr   = (float*)(ws + 3 * SL);              // -> reused as sy_b
  int*   lab_a = (int*)  (ws + 4 * SL);
  float* sy_a  = (float*)(ws + 5 * SL);
  float* sx_a  = (float*)(ws + 6 * SL);
  int*   val_a = (int*)  (ws + 7 * SL);
  float* sx_b  = (float*)(ws + 8 * SL);
  int*   val_b = (int*)  (ws + 9 * SL);
  int*   segs  = (int*)  (ws + 10 * SL);
  int*   dmax  = (int*)  (ws + 11 * SL);

  int*   peaks = (int*)rowd;                         // alias after EDT done
  int*   lab_b = (int*)dist;                         // alias after peaks done
  float* sy_b  = (float*)mpr;

  k_fg_wmma     <<<dim3(128),      dim3(256), 0, stream>>>(x, fg);
  k_rowdist     <<<dim3(8),        dim3(256), 0, stream>>>(fg, rowd);
  k_edt         <<<dim3(B_ * W_),  dim3(256), 0, stream>>>(rowd, dist);
  k_rowmax      <<<dim3(B_ * H_),  dim3(256), 0, stream>>>(dist, mpr);
  k_colmax_peaks<<<dim3(B_ * W_),  dim3(256), 0, stream>>>(mpr, dist, fg, peaks);
  k_label       <<<dim3(B_),       dim3(256), 0, stream>>>(peaks, lab_a, sy_a, sx_a, val_a, dmax);

  float *s0y = sy_a, *s0x = sx_a, *s1y = sy_b, *s1x = sx_b;
  int   *l0 = lab_a, *v0 = val_a, *l1 = lab_b, *v1 = val_b;
  for (int s = 128; s >= 1; s >>= 1) {               // 8 JFA steps -> ends in a-set
    k_jfa<<<dim3(N_ / 256), dim3(256), 0, stream>>>(s0y, s0x, l0, v0,
                                                    s1y, s1x, l1, v1, s);
    float* tf;
    int* ti;
    tf = s0y; s0y = s1y; s1y = tf;
    tf = s0x; s0x = s1x; s1x = tf;
    ti = l0;  l0  = l1;  l1  = ti;
    ti = v0;  v0  = v1;  v1  = ti;
  }

  k_segs  <<<dim3(N_ / 256),           dim3(256), 0, stream>>>(fg, l0, segs, dmax);
  k_onehot<<<dim3((B_ * G_ * HW_) / 256), dim3(256), 0, stream>>>(segs, dmax, out);
}